// RoPE_63187558858909
// MI455X (gfx1250) — compile-verified
//
#include <hip/hip_runtime.h>
#include <hip/hip_bf16.h>

// RoPE "rotation matrix * x" — but elementwise, so the output is a banded
// matrix: zero except 2 entries per row. Optimal MI455X strategy:
//   - 256 MB streaming zero-fill at HBM write bandwidth (~11 us at 23.3 TB/s)
//     via the CDNA5 async LDS->global b128 store path (ASYNCcnt-tracked,
//     zero data-VGPR cost, 512 B per wave32 instruction), with TH=NT since
//     the 256 MB output exceeds the 192 MB L2 and is never re-read on device.
//   - Tiny banded compute: 16K sincos + 0.5 MB of loads, fused in-kernel,
//     written with non-temporal b128 stores to disjoint addresses.
// WMMA is deliberately NOT used: arithmetic intensity ~0.001 flop/byte,
// no contraction dimension exists. This op lives entirely on the HBM roof.

typedef float v4f __attribute__((ext_vector_type(4)));

#ifndef D_MODEL
#define D_MODEL 2048
#endif
// log2(10000) / (D/2) = 13.287712379549449 / 1024
#define NEG_LOG2_BASE_OVER_HALF 0.012976281620653759f

#if __has_builtin(__builtin_amdgcn_global_store_async_from_lds_b128)
#define HAVE_ASYNC_STORE 1
// Builtin signature (from the round-1 diagnostic): param 0 is
// 'int __vector(4) __device__ *'  ==  int32x4 in addrspace(1) (global).
// Mirror that for the LDS-side source pointer in addrspace(3).
typedef int v4i_t __attribute__((vector_size(16)));
typedef __attribute__((address_space(1))) v4i_t gbl_v4i;
typedef __attribute__((address_space(3))) v4i_t lds_v4i;
// Store temporal-hint: TH[2:0]=1 -> NT (ISA cache-controls table).
#define CPOL_TH_NT 1
#else
#define HAVE_ASYNC_STORE 0
#endif

__global__ void rope_band_zero_kernel(const float* __restrict__ x,
                                      const int* __restrict__ m_ptr,
                                      float* __restrict__ out,
                                      long total4) {
#if HAVE_ASYNC_STORE
  // 16 bytes of zeros in LDS; every lane of every zero-store broadcasts-reads
  // this one slot (ds banks broadcast same-address reads).
  __shared__ v4f zlds;
  if (threadIdx.x == 0) {
    v4f z = {0.f, 0.f, 0.f, 0.f};
    zlds = z;
  }
  __syncthreads();  // make the LDS zeros visible before async stores read them
  lds_v4i* zsrc = (lds_v4i*)&zlds;
#endif

  const float m = (float)(*m_ptr);
  const long stride = (long)gridDim.x * blockDim.x;
  long v = (long)blockIdx.x * blockDim.x + threadIdx.x;

  for (; v < total4; v += stride) {
    const long row = v >> 9;           // 512 float4-groups per 2048-wide row
    const int g = (int)(v & 511);      // float4-group within the row
    const int i = (int)(row & (D_MODEL - 1));  // row index within the matrix
    float* dst = out + (v << 2);

    if (g == (i >> 2)) {
      // Band group: holds columns {2p, 2p+1}, p = i>>1.
      const int p = i >> 1;
      const float ang = m * exp2f(-(float)p * NEG_LOG2_BASE_OVER_HALF);
      float s, c;
      sincosf(ang, &s, &c);
      const float2 x2 = *(const float2*)(x + (row << 11) + (p << 1));
      float a, b;
      if ((i & 1) == 0) {  // even row of the 2x2 block: ( c, -s)
        a = c * x2.x;
        b = -s * x2.y;
      } else {             // odd row of the 2x2 block:  ( s,  c)
        a = s * x2.x;
        b = c * x2.y;
      }
      v4f o = {0.f, 0.f, 0.f, 0.f};
      if (i & 2) { o.z = a; o.w = b; }   // pair sits in upper half of group
      else       { o.x = a; o.y = b; }   // pair sits in lower half of group
      __builtin_nontemporal_store(o, (v4f*)dst);
    } else {
      // Bulk zero-fill: the HBM-bound 256 MB stream.
#if HAVE_ASYNC_STORE
      // CDNA5 async LDS->global store (ASYNCcnt): 16B/lane, 512B/wave-inst,
      // non-temporal so the write stream doesn't cycle the 192 MB L2.
      // Disjoint addresses from the band stores; S_ENDPGM drains ASYNCcnt.
      __builtin_amdgcn_global_store_async_from_lds_b128(
          (gbl_v4i*)dst, zsrc, /*offset=*/0, /*cpol=*/CPOL_TH_NT);
#else
      v4f z = {0.f, 0.f, 0.f, 0.f};
      __builtin_nontemporal_store(z, (v4f*)dst);
#endif
    }
  }
}

extern "C" void kernel_launch(void* const* d_in, const int* in_sizes, int n_in,
                              void* d_out, int out_size, void* d_ws, size_t ws_size,
                              hipStream_t stream) {
  const float* x = (const float*)d_in[0];   // (B, D, D) f32
  const int* m_ptr = (const int*)d_in[1];   // scalar m (device-side)
  float* out = (float*)d_out;               // (B, D, D) f32

  const long total4 = (long)out_size >> 2;  // 16.7M float4 groups
  const int threads = 256;                  // 8 waves / workgroup on wave32
  long blocks_l = (total4 + (long)threads * 8 - 1) / ((long)threads * 8);
  if (blocks_l < 1) blocks_l = 1;
  if (blocks_l > 16384) blocks_l = 16384;
  const int blocks = (int)blocks_l;         // ~8 b128 stores per thread

  rope_band_zero_kernel<<<blocks, threads, 0, stream>>>(x, m_ptr, out, total4);
}